// StabilisedStopGradientDPF_36086315221165
// MI455X (gfx1250) — compile-verified
//
#include <hip/hip_runtime.h>
#include <math.h>

// Problem constants (match reference)
#define N_PART 512
#define D_ST   8
#define E_OBS  4
#define B_BATCH 8
#define T_STEPS 24
#define NWAVE  (N_PART / 32)

// Q=0.1 -> 1/Q = 10, -0.5/Q = -5 ; R=0.5 -> -0.5/R = -1
// Cobs = -0.5*E*(log(2pi)+log(R)) = -2*(1.8378770664 - 0.6931471806)
#define C_OBS (-2.2894597716f)

typedef float v2f __attribute__((ext_vector_type(2)));
typedef float v8f __attribute__((ext_vector_type(8)));

// ---------------- Tensor Data Mover (async noise prefetch) ----------------
#if defined(__has_builtin)
#  if __has_builtin(__builtin_amdgcn_tensor_load_to_lds) && __has_builtin(__builtin_amdgcn_s_wait_tensorcnt)
#    define HAVE_TDM 1
#  endif
#endif
#ifndef HAVE_TDM
#  define HAVE_TDM 0
#endif

#if HAVE_TDM
typedef unsigned int u32x4 __attribute__((ext_vector_type(4)));
typedef int i32x4 __attribute__((ext_vector_type(4)));
typedef int i32x8 __attribute__((ext_vector_type(8)));

// DMA one contiguous 4096-float (16KB) row: the [N,D] noise tile of one step.
__device__ __forceinline__ void tdm_prefetch_noise(const float* gsrc, unsigned lds_off) {
  unsigned long long ga = (unsigned long long)gsrc;
  u32x4 g0;
  g0.x = 1u;                                                    // count=1 (valid D#)
  g0.y = lds_off;                                               // LDS byte address
  g0.z = (unsigned)ga;                                          // global_addr[31:0]
  g0.w = (unsigned)((ga >> 32) & 0x01FFFFFFull) | 0x80000000u;  // ga[56:32] | type=2
  i32x8 g1;
  g1[0] = 0x00020000;             // workgroup_mask=0, data_size=2 (4 bytes)
  g1[1] = (int)(4096u << 16);     // tensor_dim0[15:0] = 4096
  g1[2] = 0x00010000;             // tensor_dim0 hi = 0 | tensor_dim1 = 1
  g1[3] = (int)(4096u << 16);     // tile_dim0 = 4096
  g1[4] = 0;                      // tile_dim1 = 0 (unused), tile_dim2 = 0
  g1[5] = 4096;                   // tensor_dim0_stride
  g1[6] = 0;
  g1[7] = 0;
  i32x4 z4 = {0, 0, 0, 0};
#if __clang_major__ >= 23
  i32x8 z8 = {0, 0, 0, 0, 0, 0, 0, 0};
  __builtin_amdgcn_tensor_load_to_lds(g0, g1, z4, z4, z8, 0);
#else
  __builtin_amdgcn_tensor_load_to_lds(g0, g1, z4, z4, 0);
#endif
}
#endif

// ---------------- wave32 reductions (no barriers) ----------------
__device__ __forceinline__ float wsum(float v) {
#pragma unroll
  for (int o = 16; o >= 1; o >>= 1) v += __shfl_xor(v, o, 32);
  return v;
}
__device__ __forceinline__ float wmax(float v) {
#pragma unroll
  for (int o = 16; o >= 1; o >>= 1) v = fmaxf(v, __shfl_xor(v, o, 32));
  return v;
}

// log-normalize a per-thread value over the whole block (512 = 16 waves)
__device__ __forceinline__ float normalize_log(float wr, float* part) {
  const int lane = threadIdx.x & 31, wv = threadIdx.x >> 5;
  float m = wmax(wr);
  if (lane == 0) part[wv] = m;
  __syncthreads();
  float gm = part[0];
#pragma unroll
  for (int i = 1; i < NWAVE; ++i) gm = fmaxf(gm, part[i]);
  __syncthreads();
  float s = wsum(__expf(wr - gm));
  if (lane == 0) part[wv] = s;
  __syncthreads();
  float gs = part[0];
#pragma unroll
  for (int i = 1; i < NWAVE; ++i) gs += part[i];
  __syncthreads();
  return wr - (gm + __logf(gs));
}

// weighted mean over particles: out[d] = sum_n p_n * x_n[d]
__device__ __forceinline__ void out_mean(const float* x, float p, float* part8, float* outp) {
  const int lane = threadIdx.x & 31, wv = threadIdx.x >> 5, tid = threadIdx.x;
#pragma unroll
  for (int d = 0; d < D_ST; ++d) {
    float r = wsum(p * x[d]);
    if (lane == 0) part8[wv * D_ST + d] = r;
  }
  __syncthreads();
  if (tid < D_ST) {
    float acc = 0.f;
#pragma unroll
    for (int w = 0; w < NWAVE; ++w) acc += part8[w * D_ST + tid];
    outp[tid] = acc;
  }
  __syncthreads();
}

// -0.5*sum((y - C s)^2)/R + Cobs, R=0.5 -> coefficient -1
__device__ __forceinline__ float obs_ll(const float* st, const float* sC, const float* yd) {
  float acc = 0.f;
#pragma unroll
  for (int e = 0; e < E_OBS; ++e) {
    float m = 0.f;
#pragma unroll
    for (int d = 0; d < D_ST; ++d) m = fmaf(sC[e * D_ST + d], st[d], m);
    float df = yd[e] - m;
    acc = fmaf(df, df, acc);
  }
  return -acc + C_OBS;
}

__global__ __launch_bounds__(N_PART, 1) void dpf_kernel(
    const float* __restrict__ data,        // [T+1,B,E]
    const float* __restrict__ init_state,  // [B,N,D]
    const float* __restrict__ noise,       // [T,B,N,D]
    const float* __restrict__ offsets,     // [T,B]
    const float* __restrict__ Amat,        // [D,D]
    const float* __restrict__ Cmat,        // [E,D]
    float* __restrict__ out)               // [T+1,B,D]
{
  const int b    = blockIdx.x;
  const int tid  = threadIdx.x;
  const int lane = tid & 31;
  const int wv   = tid >> 5;
  const int lrow = lane & 15;
  const int kh   = lane >> 4;

  __shared__ __align__(16) float s_state[N_PART][D_ST];  // state, overwritten by `new`
  __shared__ __align__(16) float s_mean [N_PART][D_ST];  // A @ resampled state
#if HAVE_TDM
  __shared__ __align__(16) float s_noise[N_PART][D_ST];  // TDM destination
#endif
  __shared__ __align__(16) float s_rnum[N_PART];
  __shared__ __align__(16) float s_rden[N_PART];
  __shared__ float s_cum [N_PART];
  __shared__ float s_part[NWAVE];
  __shared__ float s_part8[NWAVE * D_ST];
  __shared__ float s_A[D_ST * D_ST];
  __shared__ float s_C[E_OBS * D_ST];
  __shared__ float s_y[E_OBS];

#if HAVE_TDM
  const unsigned noise_lds = (unsigned)(unsigned long long)(void*)&s_noise[0][0];
  if (tid == 0) {  // prefetch noise for step t=1 right away
    tdm_prefetch_noise(noise + (size_t)b * N_PART * D_ST, noise_lds);
  }
#endif

  if (tid < D_ST * D_ST) s_A[tid] = Amat[tid];
  if (tid < E_OBS * D_ST) s_C[tid] = Cmat[tid];
  if (tid < E_OBS) s_y[tid] = data[b * E_OBS + tid];
  {
    const float4* src = (const float4*)(init_state + (size_t)b * N_PART * D_ST);
    float4* dst = (float4*)&s_state[0][0];
    dst[tid * 2 + 0] = src[tid * 2 + 0];
    dst[tid * 2 + 1] = src[tid * 2 + 1];
  }
  __syncthreads();

  // ---- t = 0: bootstrap ----
  float wcur;  // this thread's particle log-weight (carried in register)
  {
    float stt[D_ST];
#pragma unroll
    for (int d = 0; d < D_ST; ++d) stt[d] = s_state[tid][d];
    float w0 = obs_ll(stt, s_C, s_y);
    wcur = normalize_log(w0, s_part);
    out_mean(stt, __expf(wcur), s_part8, out + (size_t)b * D_ST);
  }

  // ---- time loop ----
  for (int t = 1; t <= T_STEPS; ++t) {
    if (tid < E_OBS) s_y[tid] = data[(size_t)t * B_BATCH * E_OBS + b * E_OBS + tid];
    const float off = offsets[(size_t)(t - 1) * B_BATCH + b];

    // cumsum(exp(w)): wave shfl scan + cross-wave offsets (2 barriers)
    {
      float v = __expf(wcur);
#pragma unroll
      for (int o = 1; o < 32; o <<= 1) {
        float u = __shfl_up(v, o, 32);
        if (lane >= o) v += u;
      }
      if (lane == 31) s_part[wv] = v;
      __syncthreads();
      float offw = 0.f;
#pragma unroll
      for (int i = 0; i < NWAVE; ++i) offw += (i < wv) ? s_part[i] : 0.f;
      float cum = fminf(v + offw, 1.0f);
      if (tid == N_PART - 1) cum = 1.0f;
      s_cum[tid] = cum;
      __syncthreads();
    }

    // systematic resample: searchsorted(cum*N, tid+off, side='left')
    const float target = (float)tid + off;
    int lo = 0, hi = N_PART;
    while (lo < hi) {
      int mid = (lo + hi) >> 1;
      if (s_cum[mid] * (float)N_PART >= target) hi = mid; else lo = mid + 1;
    }
    const int idx = (lo < N_PART) ? lo : (N_PART - 1);

    float rx[D_ST];
#pragma unroll
    for (int d = 0; d < D_ST; ++d) rx[d] = s_state[idx][d];

#if HAVE_TDM
    __builtin_amdgcn_s_wait_tensorcnt(0);  // noise tile for this step has landed
#endif
    __syncthreads();  // all gathers done -> s_state free; noise visible to all

    // propagate: mean_old = A @ rx ; new = mean_old + noise
    float mo[D_ST], nw[D_ST];
#pragma unroll
    for (int k = 0; k < D_ST; ++k) {
      float m = 0.f;
#pragma unroll
      for (int d = 0; d < D_ST; ++d) m = fmaf(s_A[k * D_ST + d], rx[d], m);
      mo[k] = m;
    }
#if HAVE_TDM
    const float* nzp = &s_noise[tid][0];
    const float nz[D_ST] = {nzp[0], nzp[1], nzp[2], nzp[3], nzp[4], nzp[5], nzp[6], nzp[7]};
#else
    const float4* np4 = (const float4*)(noise +
        (((size_t)(t - 1) * B_BATCH + b) * N_PART + tid) * D_ST);
    float4 n0 = np4[0], n1 = np4[1];
    const float nz[D_ST] = {n0.x, n0.y, n0.z, n0.w, n1.x, n1.y, n1.z, n1.w};
#endif
    float sq = 0.f;
#pragma unroll
    for (int k = 0; k < D_ST; ++k) {
      nw[k] = mo[k] + nz[k];
      sq = fmaf(nw[k], nw[k], sq);
      s_mean [tid][k] = mo[k];
      s_state[tid][k] = nw[k];  // `new` becomes the carried state (A operand)
    }
    const float ob = obs_ll(nw, s_C, s_y);
    s_rden[tid] = -5.0f * sq;       // -0.5/Q * ||new_i||^2
    s_rnum[tid] = -5.0f * sq + ob;  // + obs loglik(i)
    __syncthreads();

#if HAVE_TDM
    if (tid == 0 && t < T_STEPS) {  // overlap next step's noise DMA with WMMA phase
      tdm_prefetch_noise(noise + ((size_t)t * B_BATCH + b) * N_PART * D_ST, noise_lds);
    }
#endif

    // ---- pairwise cross term via WMMA f32 16x16x4 (K=8 as 2 chained) ----
    const int j0 = (wv * 2 + 0) * 16;
    const int j1 = (wv * 2 + 1) * 16;
    v2f B00, B01, B10, B11;
    B00.x = s_mean[j0 + lrow][2 * kh + 0];     B00.y = s_mean[j0 + lrow][2 * kh + 1];
    B01.x = s_mean[j0 + lrow][4 + 2 * kh + 0]; B01.y = s_mean[j0 + lrow][4 + 2 * kh + 1];
    B10.x = s_mean[j1 + lrow][2 * kh + 0];     B10.y = s_mean[j1 + lrow][2 * kh + 1];
    B11.x = s_mean[j1 + lrow][4 + 2 * kh + 0]; B11.y = s_mean[j1 + lrow][4 + 2 * kh + 1];

    float mn0 = -INFINITY, sn0 = 0.f, md0 = -INFINITY, sd0 = 0.f;
    float mn1 = -INFINITY, sn1 = 0.f, md1 = -INFINITY, sd1 = 0.f;
    const float4* rn4 = (const float4*)s_rnum;
    const float4* rd4 = (const float4*)s_rden;

    auto lse_tile = [&](const v8f& g, const float* rn, const float* rd,
                        float& mn, float& sn, float& md, float& sd) {
      float xn[8], xd[8];
#pragma unroll
      for (int v = 0; v < 8; ++v) {
        float gv = 10.0f * g[v];
        xn[v] = gv + rn[v];
        xd[v] = gv + rd[v];
      }
      float mtn = xn[0], mtd = xd[0];
#pragma unroll
      for (int v = 1; v < 8; ++v) { mtn = fmaxf(mtn, xn[v]); mtd = fmaxf(mtd, xd[v]); }
      float stn = 0.f, std_ = 0.f;
#pragma unroll
      for (int v = 0; v < 8; ++v) { stn += __expf(xn[v] - mtn); std_ += __expf(xd[v] - mtd); }
      float nmn = fmaxf(mn, mtn);
      sn = sn * __expf(mn - nmn) + stn * __expf(mtn - nmn); mn = nmn;
      float nmd = fmaxf(md, mtd);
      sd = sd * __expf(md - nmd) + std_ * __expf(mtd - nmd); md = nmd;
    };

    for (int it = 0; it < N_PART / 16; ++it) {
      const int ib = it * 16;
      v2f A0, A1;
      A0.x = s_state[ib + lrow][2 * kh + 0];     A0.y = s_state[ib + lrow][2 * kh + 1];
      A1.x = s_state[ib + lrow][4 + 2 * kh + 0]; A1.y = s_state[ib + lrow][4 + 2 * kh + 1];

      v8f g0 = {};
      g0 = __builtin_amdgcn_wmma_f32_16x16x4_f32(false, A0, false, B00, (short)0, g0, false, false);
      g0 = __builtin_amdgcn_wmma_f32_16x16x4_f32(false, A1, false, B01, (short)0, g0, false, false);
      v8f g1 = {};
      g1 = __builtin_amdgcn_wmma_f32_16x16x4_f32(false, A0, false, B10, (short)0, g1, false, false);
      g1 = __builtin_amdgcn_wmma_f32_16x16x4_f32(false, A1, false, B11, (short)0, g1, false, false);

      const int r4 = it * 4 + kh * 2;  // rows i = ib + 8*kh + v
      float4 ra = rn4[r4], rb = rn4[r4 + 1];
      float4 da = rd4[r4], db = rd4[r4 + 1];
      const float rn[8] = {ra.x, ra.y, ra.z, ra.w, rb.x, rb.y, rb.z, rb.w};
      const float rd[8] = {da.x, da.y, da.z, da.w, db.x, db.y, db.z, db.w};

      lse_tile(g0, rn, rd, mn0, sn0, md0, sd0);
      lse_tile(g1, rn, rd, mn1, sn1, md1, sd1);
    }

    // combine half-wave row partials per column j
    auto comb = [&](float& m, float& s) {
      float mo_ = __shfl_xor(m, 16, 32);
      float so_ = __shfl_xor(s, 16, 32);
      float nm = fmaxf(m, mo_);
      s = s * __expf(m - nm) + so_ * __expf(mo_ - nm);
      m = nm;
    };
    comb(mn0, sn0); comb(md0, sd0); comb(mn1, sn1); comb(md1, sd1);

    // lane L of wave wv produced column wv*32+L == tid: weight stays in-register
    const float w0v = (mn0 + __logf(sn0)) - (md0 + __logf(sd0));
    const float w1v = (mn1 + __logf(sn1)) - (md1 + __logf(sd1));
    const float wr = (lane < 16) ? w0v : w1v;

    wcur = normalize_log(wr, s_part);
    const float p = __expf(wcur);
    out_mean(nw, p, s_part8, out + ((size_t)t * B_BATCH + b) * D_ST);
  }
}

extern "C" void kernel_launch(void* const* d_in, const int* in_sizes, int n_in,
                              void* d_out, int out_size, void* d_ws, size_t ws_size,
                              hipStream_t stream) {
  (void)in_sizes; (void)n_in; (void)out_size; (void)d_ws; (void)ws_size;
  const float* data       = (const float*)d_in[0];
  const float* init_state = (const float*)d_in[1];
  const float* prop_noise = (const float*)d_in[2];
  const float* offsets    = (const float*)d_in[3];
  const float* A          = (const float*)d_in[4];
  const float* C          = (const float*)d_in[5];
  float* out = (float*)d_out;
  dpf_kernel<<<B_BATCH, N_PART, 0, stream>>>(data, init_state, prop_noise, offsets, A, C, out);
}